// SimpleGNN_53317724013383
// MI455X (gfx1250) — compile-verified
//
#include <hip/hip_runtime.h>
#include <hip/hip_bf16.h>

typedef __attribute__((ext_vector_type(2))) float v2f;
typedef __attribute__((ext_vector_type(8))) float v8f;

// ---------------------------------------------------------------------------
// Kernel 1: deg init (self-loop contributes 1 to every node)
// ---------------------------------------------------------------------------
__global__ void k_deg_init(float* __restrict__ deg, int n) {
    int i = blockIdx.x * blockDim.x + threadIdx.x;
    if (i < n) deg[i] = 1.0f;
}

// ---------------------------------------------------------------------------
// Kernel 2: deg accumulation over edge destinations
// ---------------------------------------------------------------------------
__global__ void k_deg_accum(const int* __restrict__ dst, float* __restrict__ deg, int e) {
    int i = blockIdx.x * blockDim.x + threadIdx.x;
    if (i < e) atomicAdd(&deg[dst[i]], 1.0f);
}

// ---------------------------------------------------------------------------
// Kernel 3: dinv = rsqrt(deg), in place
// ---------------------------------------------------------------------------
__global__ void k_dinv(float* __restrict__ deg, int n) {
    int i = blockIdx.x * blockDim.x + threadIdx.x;
    if (i < n) deg[i] = rsqrtf(deg[i]);
}

// ---------------------------------------------------------------------------
// Kernel 4: h = x @ W1 via V_WMMA_F32_16X16X4_F32 (K=2 zero-padded to 4).
// One wave32 per 16-node tile. Also initializes agg1 with the self-loop
// term h[i]*dinv[i]^2 so the edge kernel only needs atomics for real edges.
//
// A (16x4 f32): lanes 0-15 hold M=lane, V0=K0,V1=K1; lanes 16-31 hold K2,K3 (zero pad)
// B (4x16 f32): lanes 0-15 hold N=lane, V0=K0,V1=K1; lanes 16-31 hold K2,K3 (zero pad)
// D (16x16 f32): lane<16: VGPR g = D[g][lane]; lane>=16: D[g+8][lane-16]
// ---------------------------------------------------------------------------
__global__ void k_h_wmma(const float* __restrict__ x, const float* __restrict__ W1,
                         const float* __restrict__ dinv,
                         float* __restrict__ h, float* __restrict__ agg1, int n) {
    int wave = (blockIdx.x * blockDim.x + threadIdx.x) >> 5;
    int lane = threadIdx.x & 31;
    int row0 = wave * 16;
    if (row0 >= n) return;  // wave-uniform: EXEC is all-ones (or all-zero) at WMMA

    int m = lane & 15;
    v2f a = {0.0f, 0.0f};
    v2f b = {0.0f, 0.0f};
    if (lane < 16) {                 // lanes 16-31 carry K=2,3 -> zero padding
        int r = row0 + m;
        if (r < n) {
            a.x = x[r * 2 + 0];      // K=0
            a.y = x[r * 2 + 1];      // K=1
        }
        b.x = W1[0 * 16 + m];        // W1 row K=0, col N=m
        b.y = W1[1 * 16 + m];        // W1 row K=1
    }

    v8f c = {};
    c = __builtin_amdgcn_wmma_f32_16x16x4_f32(
            /*neg_a=*/false, a, /*neg_b=*/false, b,
            /*c_mod=*/(short)0, c, /*reuse_a=*/false, /*reuse_b=*/false);

    int rbase = row0 + ((lane >> 4) << 3);   // +8 rows for upper half-wave
    int col   = lane & 15;

    if (row0 + 16 <= n) {
        // Full tile: branch-free. Hoist the 8 dinv loads so they clause
        // together, then issue 16 coalesced stores (2 rows x 16 cols per g).
        float di[8];
#pragma unroll
        for (int g = 0; g < 8; ++g) di[g] = dinv[rbase + g];
#pragma unroll
        for (int g = 0; g < 8; ++g) {
            int row = rbase + g;
            float v = c[g];
            h[row * 16 + col]    = v;
            agg1[row * 16 + col] = v * di[g] * di[g];
        }
    } else {
        // Possible ragged last tile (n % 16 != 0): guarded path.
#pragma unroll
        for (int g = 0; g < 8; ++g) {
            int row = rbase + g;
            if (row < n) {
                float v  = c[g];
                float di = dinv[row];
                h[row * 16 + col]    = v;
                agg1[row * 16 + col] = v * di * di;
            }
        }
    }
}

// ---------------------------------------------------------------------------
// Kernel 5: layer-1 edge aggregation. 16 lanes cooperate on one edge
// (lane f handles feature f) -> coalesced 64B gathers and scatter-atomics.
// ---------------------------------------------------------------------------
__global__ void k_agg1(const int* __restrict__ src, const int* __restrict__ dst,
                       const float* __restrict__ h, const float* __restrict__ dinv,
                       float* __restrict__ agg1, int e) {
    long long t = (long long)blockIdx.x * blockDim.x + threadIdx.x;
    long long ed = t >> 4;
    int f = (int)(t & 15);
    if (ed >= (long long)e) return;
    int s = src[ed];
    int d = dst[ed];
    float norm = dinv[s] * dinv[d];
    float v = h[s * 16 + f] * norm;
    atomicAdd(&agg1[d * 16 + f], v);
}

// ---------------------------------------------------------------------------
// Kernel 6: per node: a = relu(agg1 + b1); s = dot(a, W2); agg2 init with
// self-loop term s*dinv^2.
// ---------------------------------------------------------------------------
__global__ void k_layer2_node(const float* __restrict__ agg1, const float* __restrict__ b1,
                              const float* __restrict__ W2, const float* __restrict__ dinv,
                              float* __restrict__ s, float* __restrict__ agg2, int n) {
    int i = blockIdx.x * blockDim.x + threadIdx.x;
    if (i >= n) return;
    const float4* a4 = (const float4*)(agg1 + i * 16);
    float acc = 0.0f;
#pragma unroll
    for (int q = 0; q < 4; ++q) {
        float4 v = a4[q];
        int f = q * 4;
        acc += fmaxf(v.x + b1[f + 0], 0.0f) * W2[f + 0];
        acc += fmaxf(v.y + b1[f + 1], 0.0f) * W2[f + 1];
        acc += fmaxf(v.z + b1[f + 2], 0.0f) * W2[f + 2];
        acc += fmaxf(v.w + b1[f + 3], 0.0f) * W2[f + 3];
    }
    float di = dinv[i];
    s[i] = acc;
    agg2[i] = acc * di * di;   // self-loop contribution for layer 2
}

// ---------------------------------------------------------------------------
// Kernel 7: layer-2 edge aggregation (scalar per edge)
// ---------------------------------------------------------------------------
__global__ void k_agg2(const int* __restrict__ src, const int* __restrict__ dst,
                       const float* __restrict__ s, const float* __restrict__ dinv,
                       float* __restrict__ agg2, int e) {
    int i = blockIdx.x * blockDim.x + threadIdx.x;
    if (i >= e) return;
    int a = src[i];
    int d = dst[i];
    atomicAdd(&agg2[d], s[a] * dinv[a] * dinv[d]);
}

// ---------------------------------------------------------------------------
// Kernel 8: per-node sigmoid(agg2 + b2)
// ---------------------------------------------------------------------------
__global__ void k_score(const float* __restrict__ agg2, const float* __restrict__ b2,
                        float* __restrict__ score, int n) {
    int i = blockIdx.x * blockDim.x + threadIdx.x;
    if (i >= n) return;
    float z = agg2[i] + b2[0];
    score[i] = 1.0f / (1.0f + __expf(-z));
}

// ---------------------------------------------------------------------------
// Kernel 9: per-edge gather of source-node score
// ---------------------------------------------------------------------------
__global__ void k_gather(const int* __restrict__ src, const float* __restrict__ score,
                         float* __restrict__ out, int e) {
    int i = blockIdx.x * blockDim.x + threadIdx.x;
    if (i < e) out[i] = score[src[i]];
}

// ---------------------------------------------------------------------------

static inline int cdiv(long long a, int b) { return (int)((a + b - 1) / b); }

extern "C" void kernel_launch(void* const* d_in, const int* in_sizes, int n_in,
                              void* d_out, int out_size, void* d_ws, size_t ws_size,
                              hipStream_t stream) {
    const float* x  = (const float*)d_in[0];   // [N,2]
    const float* W1 = (const float*)d_in[1];   // [2,16]
    const float* b1 = (const float*)d_in[2];   // [16]
    const float* W2 = (const float*)d_in[3];   // [16,1]
    const float* b2 = (const float*)d_in[4];   // [1]
    const int*   ei = (const int*)d_in[5];     // [2,E]

    const int n = in_sizes[0] / 2;             // 131072
    const int e = in_sizes[5] / 2;             // 4194304
    const int* src = ei;
    const int* dst = ei + e;

    // Workspace layout (all offsets 256B aligned); fully initialized each call.
    char* ws = (char*)d_ws;
    size_t off = 0;
    float* dinv  = (float*)(ws + off); off += (size_t)n * sizeof(float);        // deg -> dinv
    off = (off + 255) & ~(size_t)255;
    float* h     = (float*)(ws + off); off += (size_t)n * 16 * sizeof(float);
    off = (off + 255) & ~(size_t)255;
    float* agg1  = (float*)(ws + off); off += (size_t)n * 16 * sizeof(float);
    off = (off + 255) & ~(size_t)255;
    float* sbuf  = (float*)(ws + off); off += (size_t)n * sizeof(float);
    off = (off + 255) & ~(size_t)255;
    float* agg2  = (float*)(ws + off); off += (size_t)n * sizeof(float);
    off = (off + 255) & ~(size_t)255;
    float* score = (float*)(ws + off); off += (size_t)n * sizeof(float);
    (void)ws_size; (void)n_in;

    float* out = (float*)d_out;                // [E]

    const int B = 256;

    // degree (with self-loops) -> dinv
    k_deg_init <<<cdiv(n, B), B, 0, stream>>>(dinv, n);
    k_deg_accum<<<cdiv(e, B), B, 0, stream>>>(dst, dinv, e);
    k_dinv     <<<cdiv(n, B), B, 0, stream>>>(dinv, n);

    // h = x @ W1 (WMMA f32 16x16x4), agg1 pre-seeded with self-loop term
    {
        int waves = cdiv(n, 16);               // one wave32 per 16-row tile
        int wavesPerBlock = B / 32;
        k_h_wmma<<<cdiv(waves, wavesPerBlock), B, 0, stream>>>(x, W1, dinv, h, agg1, n);
    }

    // layer-1 edge aggregation: 16 lanes per edge
    k_agg1<<<cdiv((long long)e * 16, B), B, 0, stream>>>(src, dst, h, dinv, agg1, e);

    // relu + dot(W2) per node, agg2 seeded with self-loop term
    k_layer2_node<<<cdiv(n, B), B, 0, stream>>>(agg1, b1, W2, dinv, sbuf, agg2, n);

    // layer-2 edge aggregation
    k_agg2<<<cdiv(e, B), B, 0, stream>>>(src, dst, sbuf, dinv, agg2, e);

    // sigmoid per node, then per-edge gather
    k_score <<<cdiv(n, B), B, 0, stream>>>(agg2, b2, score, n);
    k_gather<<<cdiv(e, B), B, 0, stream>>>(src, score, out, e);
}